// PopulationGNNClassifier_73873437491517
// MI455X (gfx1250) — compile-verified
//
#include <hip/hip_runtime.h>
#include <hip/hip_bf16.h>

// Problem constants (from the reference)
#define NN   50000
#define EE   800000
#define EP   (EE + NN)       // edges incl. self loops
#define FIN  256
#define HH   4
#define CC   64
#define HC   256
#define LEAKK 0.2f

#define LDS_STRIDE 260       // 256 + 4 DWORD TDM pad -> conflict-free ds_load

typedef float v2f __attribute__((ext_vector_type(2)));
typedef float v8f __attribute__((ext_vector_type(8)));
typedef unsigned u32x4 __attribute__((ext_vector_type(4)));
typedef int i32x4 __attribute__((ext_vector_type(4)));
typedef int i32x8 __attribute__((ext_vector_type(8)));

// ---- order-preserving float <-> uint encoding for atomic segment-max ----
__device__ __forceinline__ unsigned enc_f(float f) {
    unsigned u = __float_as_uint(f);
    return (u & 0x80000000u) ? ~u : (u | 0x80000000u);
}
__device__ __forceinline__ float dec_f(unsigned e) {
    return (e & 0x80000000u) ? __uint_as_float(e & 0x7FFFFFFFu)
                             : __uint_as_float(~e);
}

// ---------------------------------------------------------------------------
// WMMA fp32 GEMM:  Out[NN x HC] = A[NN x K] * W[K x HC]
// One block = one 16-row M strip. The 16x256 fp32 A strip (16 KB, contiguous)
// is staged into LDS by the Tensor Data Mover (one tensor_load_to_lds issued
// by wave 0), with TDM LDS-padding of 4 DWORDs per 256-DWORD row so rows sit
// 260 DWORDs apart -> conflict-free ds_load_b64 for the WMMA A operand.
// Each of the 8 waves computes two 16x16 N tiles (reusing its A fragment).
// A 16x4 layout (ISA 7.12.2): lane&15 = row M, K = 2*(lane>>4) + vgpr.
// B 4x16 mirrored: lane&15 = N col, K = 2*(lane>>4) + vgpr.
// C/D: col = lane&15, row = vgpr + 8*(lane>>4).
// ---------------------------------------------------------------------------
__global__ __launch_bounds__(256) void gat_gemm_wmma_f32(
    const float* __restrict__ A, const float* __restrict__ W,
    float* __restrict__ Out, int K)
{
    __shared__ float shA[16 * LDS_STRIDE];   // 16 rows, padded stride

    const int wave = threadIdx.x >> 5;
    const int lane = threadIdx.x & 31;
    const int tileM = blockIdx.x;            // NN/16 strips
    const int half = lane >> 4;
    const int mn   = lane & 15;

    if (threadIdx.x < 32) {
        // ---- TDM: stage contiguous 16x256 f32 strip into LDS with padding ----
        unsigned lds_off = (unsigned)(uintptr_t)(&shA[0]);
        unsigned long long ga =
            (unsigned long long)(uintptr_t)(A + (size_t)tileM * 16 * K);
        const unsigned td0 = 16u * (unsigned)K;          // 4096 elements (f32)

        u32x4 g0;
        g0[0] = 1u;                                      // count=1, user mode
        g0[1] = lds_off;                                 // lds_addr (bytes)
        g0[2] = (unsigned)(ga & 0xFFFFFFFFu);            // global_addr[31:0]
        g0[3] = (unsigned)((ga >> 32) & 0x01FFFFFFu)     // global_addr[56:32]
              | 0x80000000u;                             // type=2 ("image")

        i32x8 g1;
        g1[0] = (int)(0x00020000u                        // data_size=2 (4B)
              | (1u << 20)                               // pad_enable
              | (7u << 22)                               // pad_interval: 256 DW
              | (3u << 25));                             // pad_amount:   4 DW
        g1[1] = (int)((td0 & 0xFFFFu) << 16);            // tensor_dim0[15:0]
        g1[2] = (int)(((td0 >> 16) & 0xFFFFu)            // tensor_dim0[31:16]
              | (1u << 16));                             // tensor_dim1 = 1
        g1[3] = (int)((td0 & 0xFFFFu) << 16);            // tile_dim0 = 4096
        g1[4] = 0;                                       // tile_dim1/2 unused
        g1[5] = (int)td0;                                // tensor_dim0_stride
        g1[6] = 0;
        g1[7] = 0;

        i32x4 gz4 = {0, 0, 0, 0};
        i32x8 gz8 = {0, 0, 0, 0, 0, 0, 0, 0};
        __builtin_amdgcn_tensor_load_to_lds(g0, g1, gz4, gz4, gz8, 0);
        __builtin_amdgcn_s_wait_tensorcnt(0);
    }
    __syncthreads();

    const int n0 = wave * 2;                 // two 16-col tiles per wave
    const float* Wc0 = W + n0 * 16 + mn;
    const float* Wc1 = W + (n0 + 1) * 16 + mn;
    const float* aRow = shA + mn * LDS_STRIDE + 2 * half;

    v8f acc0 = {};
    v8f acc1 = {};
    for (int k = 0; k < K; k += 4) {
        v2f a, b0, b1;
        a.x = aRow[k];
        a.y = aRow[k + 1];
        const size_t kr = (size_t)(k + 2 * half) * HC;
        b0.x = Wc0[kr];
        b0.y = Wc0[kr + HC];
        b1.x = Wc1[kr];
        b1.y = Wc1[kr + HC];
        acc0 = __builtin_amdgcn_wmma_f32_16x16x4_f32(
            false, a, false, b0, (short)0, acc0, false, false);
        acc1 = __builtin_amdgcn_wmma_f32_16x16x4_f32(
            false, a, false, b1, (short)0, acc1, false, false);
    }
    float* o0 = Out + (size_t)(tileM * 16 + 8 * half) * HC + n0 * 16 + mn;
#pragma unroll
    for (int v = 0; v < 8; ++v) {
        o0[(size_t)v * HC]      = acc0[v];
        o0[(size_t)v * HC + 16] = acc1[v];
    }
}

// ---- per-(node,head) attention dots: a_src[n,h], a_dst[n,h] ----
__global__ void gat_att_node(const float* __restrict__ xs,
                             const float* __restrict__ a_s,
                             const float* __restrict__ a_d,
                             float* __restrict__ asrc, float* __restrict__ adst)
{
    int t = blockIdx.x * blockDim.x + threadIdx.x;
    if (t >= NN * HH) return;
    const int h = t & 3;
    const float4* xp = (const float4*)(xs + ((size_t)(t >> 2)) * HC + h * CC);
    const float4* sp = (const float4*)(a_s + h * CC);
    const float4* dp = (const float4*)(a_d + h * CC);
    float s1 = 0.f, s2 = 0.f;
#pragma unroll 4
    for (int i = 0; i < 16; ++i) {
        float4 xv = xp[i], sv = sp[i], dv = dp[i];
        s1 += xv.x * sv.x + xv.y * sv.y + xv.z * sv.z + xv.w * sv.w;
        s2 += xv.x * dv.x + xv.y * dv.y + xv.z * dv.z + xv.w * dv.w;
    }
    asrc[t] = s1;
    adst[t] = s2;
}

// ---- mean(edge_attr): block reduce + one atomic per block into scal[0] ----
__global__ void gat_reduce_ea(const float* __restrict__ ea, float* __restrict__ scal)
{
    __shared__ float sm[256];
    size_t t = blockIdx.x * (size_t)blockDim.x + threadIdx.x;
    float v = 0.f;
    for (size_t i = t; i < (size_t)EE; i += (size_t)gridDim.x * blockDim.x) v += ea[i];
    sm[threadIdx.x] = v;
    __syncthreads();
    for (int s = 128; s > 0; s >>= 1) {
        if (threadIdx.x < s) sm[threadIdx.x] += sm[threadIdx.x + s];
        __syncthreads();
    }
    if (threadIdx.x == 0) atomicAdd(&scal[0], sm[0]);
}

// ---- scal[1]=ea_mean, scal[4+h]=coef1[h], scal[8+h]=coef2[h] ----
__global__ void gat_finalize_scal(const float* __restrict__ We1, const float* __restrict__ ae1,
                                  const float* __restrict__ We2, const float* __restrict__ ae2,
                                  float* __restrict__ scal)
{
    __shared__ float sm[256];
    const int t = threadIdx.x;
    sm[t] = We1[t] * ae1[t];
    __syncthreads();
    if (t < 4) { float s = 0.f; for (int i = 0; i < 64; ++i) s += sm[t * 64 + i]; scal[4 + t] = s; }
    __syncthreads();
    sm[t] = We2[t] * ae2[t];
    __syncthreads();
    if (t < 4) { float s = 0.f; for (int i = 0; i < 64; ++i) s += sm[t * 64 + i]; scal[8 + t] = s; }
    if (t == 0) scal[1] = scal[0] * (1.0f / (float)EE);
}

// ---- edge pass 1: alpha = leakyrelu(a_src[s]+a_dst[d]+ea*coef); segment max ----
__global__ void gat_edge_alpha(const int* __restrict__ ei, const float* __restrict__ ea,
                               const float* __restrict__ asrc, const float* __restrict__ adst,
                               const float* __restrict__ scal, int coefOff,
                               float* __restrict__ alpha, unsigned* __restrict__ amax)
{
    int e = blockIdx.x * blockDim.x + threadIdx.x;
    if (e >= EP) return;
    int s, d; float av;
    if (e < EE) { s = ei[e]; d = ei[EE + e]; av = ea[e]; }
    else        { s = d = e - EE;            av = scal[1]; }
    float4 as4 = *(const float4*)(asrc + (size_t)s * 4);
    float4 ad4 = *(const float4*)(adst + (size_t)d * 4);
    float4 cf  = *(const float4*)(scal + coefOff);
    float4 al;
    al.x = as4.x + ad4.x + av * cf.x;
    al.y = as4.y + ad4.y + av * cf.y;
    al.z = as4.z + ad4.z + av * cf.z;
    al.w = as4.w + ad4.w + av * cf.w;
    al.x = fmaxf(al.x, LEAKK * al.x);
    al.y = fmaxf(al.y, LEAKK * al.y);
    al.z = fmaxf(al.z, LEAKK * al.z);
    al.w = fmaxf(al.w, LEAKK * al.w);
    *(float4*)(alpha + (size_t)e * 4) = al;
    unsigned* am = amax + (size_t)d * 4;
    atomicMax(am + 0, enc_f(al.x));
    atomicMax(am + 1, enc_f(al.y));
    atomicMax(am + 2, enc_f(al.z));
    atomicMax(am + 3, enc_f(al.w));
}

// ---- edge pass 2: ex = exp(alpha - amax[d]); denom[d] += ex ----
__global__ void gat_edge_exp(const int* __restrict__ ei,
                             float* __restrict__ alpha,
                             const unsigned* __restrict__ amax,
                             float* __restrict__ denom)
{
    int e = blockIdx.x * blockDim.x + threadIdx.x;
    if (e >= EP) return;
    const int d = (e < EE) ? ei[EE + e] : (e - EE);
    float4 al = *(const float4*)(alpha + (size_t)e * 4);
    const unsigned* am = amax + (size_t)d * 4;
    float4 ex;
    ex.x = expf(al.x - dec_f(am[0]));
    ex.y = expf(al.y - dec_f(am[1]));
    ex.z = expf(al.z - dec_f(am[2]));
    ex.w = expf(al.w - dec_f(am[3]));
    *(float4*)(alpha + (size_t)e * 4) = ex;
    float* dn = denom + (size_t)d * 4;
    atomicAdd(dn + 0, ex.x);
    atomicAdd(dn + 1, ex.y);
    atomicAdd(dn + 2, ex.z);
    atomicAdd(dn + 3, ex.w);
}

// ---- edge pass 3: one wave per edge; 256 contiguous channels scatter-add ----
__global__ void gat_edge_msg(const int* __restrict__ ei,
                             const float* __restrict__ exb,
                             const float* __restrict__ denom,
                             const float* __restrict__ xs,
                             float* __restrict__ outb)
{
    long long t = blockIdx.x * (long long)blockDim.x + threadIdx.x;
    int e = (int)(t >> 5);
    if (e >= EP) return;
    const int lane = (int)(t & 31);
    int s, d;
    if (e < EE) { s = ei[e]; d = ei[EE + e]; }
    else        { s = d = e - EE; }
    const int h = lane >> 3;  // 8 lanes per head, 8 channels per lane
    const float att = exb[(size_t)e * 4 + h] / denom[(size_t)d * 4 + h];
    const float4* in = (const float4*)(xs + (size_t)s * HC + lane * 8);
    float* op = outb + (size_t)d * HC + lane * 8;
    float4 v0 = in[0], v1 = in[1];
    atomicAdd(op + 0, v0.x * att);
    atomicAdd(op + 1, v0.y * att);
    atomicAdd(op + 2, v0.z * att);
    atomicAdd(op + 3, v0.w * att);
    atomicAdd(op + 4, v1.x * att);
    atomicAdd(op + 5, v1.y * att);
    atomicAdd(op + 6, v1.z * att);
    atomicAdd(op + 7, v1.w * att);
}

// ---- bias add (+ optional ELU) ----
__global__ void gat_bias_act(const float* __restrict__ in, const float* __restrict__ b,
                             float* __restrict__ out, int do_elu)
{
    size_t t = blockIdx.x * (size_t)blockDim.x + threadIdx.x;
    if (t >= (size_t)NN * HC) return;
    float v = in[t] + b[t & (HC - 1)];
    if (do_elu) v = (v > 0.f) ? v : expm1f(v);
    out[t] = v;
}

// ---- classifier: logits = h @ Wc + bc  (HC x 2) ----
__global__ void gat_classifier(const float* __restrict__ h, const float* __restrict__ Wc,
                               const float* __restrict__ bc, float* __restrict__ out)
{
    int n = blockIdx.x * blockDim.x + threadIdx.x;
    if (n >= NN) return;
    const float4* hp = (const float4*)(h + (size_t)n * HC);
    float a0 = 0.f, a1 = 0.f;
#pragma unroll 8
    for (int i = 0; i < 64; ++i) {
        float4 hv = hp[i];
        const float4* wp = (const float4*)(Wc + i * 8);
        float4 w0 = wp[0], w1 = wp[1];
        a0 += hv.x * w0.x + hv.y * w0.z + hv.z * w1.x + hv.w * w1.z;
        a1 += hv.x * w0.y + hv.y * w0.w + hv.z * w1.y + hv.w * w1.w;
    }
    out[n * 2 + 0] = a0 + bc[0];
    out[n * 2 + 1] = a1 + bc[1];
}

extern "C" void kernel_launch(void* const* d_in, const int* in_sizes, int n_in,
                              void* d_out, int out_size, void* d_ws, size_t ws_size,
                              hipStream_t stream)
{
    (void)in_sizes; (void)n_in; (void)out_size; (void)ws_size;
    const float* x   = (const float*)d_in[0];
    const int*   ei  = (const int*)  d_in[1];
    const float* ea  = (const float*)d_in[2];
    const float* W1  = (const float*)d_in[3];
    const float* as1 = (const float*)d_in[4];
    const float* ad1 = (const float*)d_in[5];
    const float* We1 = (const float*)d_in[6];
    const float* ae1 = (const float*)d_in[7];
    const float* b1  = (const float*)d_in[8];
    const float* W2  = (const float*)d_in[9];
    const float* as2 = (const float*)d_in[10];
    const float* ad2 = (const float*)d_in[11];
    const float* We2 = (const float*)d_in[12];
    const float* ae2 = (const float*)d_in[13];
    const float* b2  = (const float*)d_in[14];
    const float* Wc  = (const float*)d_in[15];
    const float* bc  = (const float*)d_in[16];
    float* out = (float*)d_out;

    // workspace layout (fp32)
    float* xs    = (float*)d_ws;                    // N*HC   transformed features
    float* outb  = xs    + (size_t)NN * HC;         // N*HC   accumulator
    float* hb    = outb  + (size_t)NN * HC;         // N*HC   layer output
    float* alpha = hb    + (size_t)NN * HC;         // EP*H   alpha -> ex
    float* asrc  = alpha + (size_t)EP * HH;         // N*H
    float* adst  = asrc  + (size_t)NN * HH;         // N*H
    float* denom = adst  + (size_t)NN * HH;         // N*H
    unsigned* amax = (unsigned*)(denom + (size_t)NN * HH); // N*H
    float* scal  = (float*)(amax + (size_t)NN * HH);       // 16 scalars

    const int EB   = (EP + 255) / 256;
    const int MSGB = (int)(((size_t)EP * 32 + 255) / 256);
    const int GEMMB = NN / 16;                      // one 16-row strip per block
    const int NHB  = (NN * HH + 255) / 256;
    const int NHCB = (int)(((size_t)NN * HC + 255) / 256);

    // scalars: mean(edge_attr) and edge-attention coefficients
    (void)hipMemsetAsync(scal, 0, 16 * sizeof(float), stream);
    gat_reduce_ea<<<256, 256, 0, stream>>>(ea, scal);
    gat_finalize_scal<<<1, 256, 0, stream>>>(We1, ae1, We2, ae2, scal);

    // ---------------- layer 1 ----------------
    (void)hipMemsetAsync(amax,  0, (size_t)NN * HH * sizeof(unsigned), stream);
    (void)hipMemsetAsync(denom, 0, (size_t)NN * HH * sizeof(float),    stream);
    (void)hipMemsetAsync(outb,  0, (size_t)NN * HC * sizeof(float),    stream);
    gat_gemm_wmma_f32<<<GEMMB, 256, 0, stream>>>(x, W1, xs, FIN);
    gat_att_node<<<NHB, 256, 0, stream>>>(xs, as1, ad1, asrc, adst);
    gat_edge_alpha<<<EB, 256, 0, stream>>>(ei, ea, asrc, adst, scal, 4, alpha, amax);
    gat_edge_exp<<<EB, 256, 0, stream>>>(ei, alpha, amax, denom);
    gat_edge_msg<<<MSGB, 256, 0, stream>>>(ei, alpha, denom, xs, outb);
    gat_bias_act<<<NHCB, 256, 0, stream>>>(outb, b1, hb, 1);

    // ---------------- layer 2 ----------------
    (void)hipMemsetAsync(amax,  0, (size_t)NN * HH * sizeof(unsigned), stream);
    (void)hipMemsetAsync(denom, 0, (size_t)NN * HH * sizeof(float),    stream);
    (void)hipMemsetAsync(outb,  0, (size_t)NN * HC * sizeof(float),    stream);
    gat_gemm_wmma_f32<<<GEMMB, 256, 0, stream>>>(hb, W2, xs, HC);
    gat_att_node<<<NHB, 256, 0, stream>>>(xs, as2, ad2, asrc, adst);
    gat_edge_alpha<<<EB, 256, 0, stream>>>(ei, ea, asrc, adst, scal, 8, alpha, amax);
    gat_edge_exp<<<EB, 256, 0, stream>>>(ei, alpha, amax, denom);
    gat_edge_msg<<<MSGB, 256, 0, stream>>>(ei, alpha, denom, xs, outb);
    gat_bias_act<<<NHCB, 256, 0, stream>>>(outb, b2, hb, 0);

    // ---------------- classifier ----------------
    gat_classifier<<<(NN + 255) / 256, 256, 0, stream>>>(hb, Wc, bc, out);
}